// TrilinearInterpolateFeatures_80221399154757
// MI455X (gfx1250) — compile-verified
//
#include <hip/hip_runtime.h>
#include <stdint.h>

#define CCH 128            // feature channels
#define G   128            // grid resolution
#define NB  2              // batches
#define GG3 (G*G*G)        // 2097152
#define OCC_SIZE (NB*GG3)  // 4194304 ints = 16 MB
#define MAXB 16            // shift slots (safety)

// ---------------- init: occ = -1, shift = +inf, accum = 0 ----------------
__global__ void init_kernel(int* __restrict__ shift, int* __restrict__ occ,
                            float* __restrict__ accum, int n) {
    int tid = blockIdx.x * blockDim.x + threadIdx.x;
    if (tid < OCC_SIZE) occ[tid] = -1;
    if (tid < MAXB * 3) shift[tid] = 0x7fffffff;
    if (tid < n)        accum[tid] = 0.0f;
}

// ---------------- per-batch min of integer coords ----------------
__global__ void shift_kernel(const int* __restrict__ coords,
                             int* __restrict__ shift, int n) {
    int i = blockIdx.x * blockDim.x + threadIdx.x;
    if (i >= n) return;
    int b = coords[i * 4 + 0];
    atomicMin(&shift[b * 3 + 0], coords[i * 4 + 1]);
    atomicMin(&shift[b * 3 + 1], coords[i * 4 + 2]);
    atomicMin(&shift[b * 3 + 2], coords[i * 4 + 3]);
}

// ---------------- scatter point ids into dense occupancy grid ----------------
__global__ void scatter_kernel(const int* __restrict__ coords,
                               const int* __restrict__ shift,
                               int* __restrict__ occ, int n) {
    int i = blockIdx.x * blockDim.x + threadIdx.x;
    if (i >= n) return;
    int b = coords[i * 4 + 0];
    int x = coords[i * 4 + 1] - shift[b * 3 + 0];
    int y = coords[i * 4 + 2] - shift[b * 3 + 1];
    int z = coords[i * 4 + 3] - shift[b * 3 + 2];
    occ[b * GG3 + x * (G * G) + y * G + z] = i;
}

// ---------------- main query kernel: one wave32 per query ----------------
// 32 lanes cover the 128 channels as float4; lane's k=lane&7 owns corner k.
// The 8 corner feature rows are gathered with the CDNA5 async global->LDS
// copy engine (ASYNCcnt). Async loads retire in issue order, so consumption
// is software-pipelined with partial s_wait_asynccnt waits: FMA on row k
// starts as soon as that row has landed, overlapping the remaining DMAs.
__global__ __launch_bounds__(256)
void query_kernel(const float*  __restrict__ feat,
                  const float4* __restrict__ qp4,
                  const int*    __restrict__ shift,
                  const int*    __restrict__ occ,
                  float* __restrict__ qf,
                  float* __restrict__ out_idx,
                  float* __restrict__ out_w,
                  float* __restrict__ accum,
                  int M) {
    __shared__ float smem[8 * 8 * CCH];           // 8 waves x 8 corners x 128ch = 32KB
    const int lane = threadIdx.x & 31;
    const int wave = threadIdx.x >> 5;
    const int m = blockIdx.x * 8 + wave;
    if (m >= M) return;                            // wave-uniform exit, no block barriers

    // query point (broadcast load of one 16B row)
    float4 q = qp4[m];
    int b = (int)q.x;
    float fx = q.y - (float)shift[b * 3 + 0];
    float fy = q.z - (float)shift[b * 3 + 1];
    float fz = q.w - (float)shift[b * 3 + 2];
    int bx = (int)floorf(fx), by = (int)floorf(fy), bz = (int)floorf(fz);
    float rx = fx - (float)bx, ry = fy - (float)by, rz = fz - (float)bz;

    // each lane evaluates corner k = lane & 7 (4-way replicated, single instrs)
    int k  = lane & 7;
    int oi = (k >> 2) & 1, oj = (k >> 1) & 1, ok = k & 1;
    int cx = bx + oi, cy = by + oj, cz = bz + ok;
    bool inb = (cx >= 0) & (cx < G) & (cy >= 0) & (cy < G) & (cz >= 0) & (cz < G);
    int ccx = min(max(cx, 0), G - 1);
    int ccy = min(max(cy, 0), G - 1);
    int ccz = min(max(cz, 0), G - 1);
    int vox = occ[b * GG3 + ccx * (G * G) + ccy * G + ccz];
    bool valid = inb && (vox != -1);
    float w = (oi ? rx : 1.0f - rx) * (oj ? ry : 1.0f - ry) * (ok ? rz : 1.0f - rz);
    int   ivx = valid ? vox : -1;
    float wv  = valid ? w : 0.0f;

    // ---- issue async gather of the 8 corner feature rows into this wave's LDS slice ----
    // low 32 bits of a generic pointer to LDS == byte offset within the wave's
    // LDS allocation (CDNA5 aperture encoding) -> usable as async-load VDST.
    uint32_t lbase = (uint32_t)(uintptr_t)(const void*)&smem[wave * (8 * CCH)];
    float wk[8];
#pragma unroll
    for (int kk = 0; kk < 8; ++kk) {
        int   ik = __shfl(ivx, kk, 32);
        wk[kk]   = __shfl(wv,  kk, 32);
        int  sik = ik < 0 ? 0 : ik;                   // clamped gather; weight is 0 if invalid
        const float* gp = feat + (size_t)sik * CCH + lane * 4;
        uint32_t la = lbase + (uint32_t)(kk * CCH + lane * 4) * 4u;
        // each lane DMAs 16B: one instruction moves a full 512B feature row
        asm volatile("global_load_async_to_lds_b128 %0, %1, off"
                     :: "v"(la), "v"(gp) : "memory");
    }

    // per-corner side outputs + voxel-weight accumulation (lanes 0..7 only),
    // overlapped with the in-flight async DMAs
    if (lane < 8) {
        out_idx[(size_t)m * 8 + lane] = (float)ivx;   // int output stored as float
        out_w  [(size_t)m * 8 + lane] = wv;
        if (valid) atomicAdd(&accum[vox], wv);
    }

    // ---- pipelined consume: async loads retire in order, so row kk is in LDS
    // once ASYNCcnt <= 7-kk ----
    float4 acc = make_float4(0.f, 0.f, 0.f, 0.f);
#pragma unroll
    for (int kk = 0; kk < 8; ++kk) {
        asm volatile("s_wait_asynccnt %0" :: "i"(7 - kk) : "memory");
        const float4 f = *(const float4*)&smem[wave * (8 * CCH) + kk * CCH + lane * 4];
        acc.x = fmaf(wk[kk], f.x, acc.x);
        acc.y = fmaf(wk[kk], f.y, acc.y);
        acc.z = fmaf(wk[kk], f.z, acc.z);
        acc.w = fmaf(wk[kk], f.w, acc.w);
    }
    *(float4*)&qf[(size_t)m * CCH + lane * 4] = acc;
}

extern "C" void kernel_launch(void* const* d_in, const int* in_sizes, int n_in,
                              void* d_out, int out_size, void* d_ws, size_t ws_size,
                              hipStream_t stream) {
    const float* feat   = (const float*)d_in[0];   // (N, 128) f32
    const int*   coords = (const int*)  d_in[1];   // (N, 4) i32
    const float* qp     = (const float*)d_in[2];   // (M, 4) f32

    const int N = in_sizes[0] / CCH;
    const int M = in_sizes[2] / 4;

    // workspace: [0,256) shift slots, [256, 256+16MB) occupancy grid
    int* shift = (int*)d_ws;
    int* occ   = (int*)((char*)d_ws + 256);

    // outputs concatenated flat: qf (M*128), idx (M*8), w (M*8), accum (N)
    float* out   = (float*)d_out;
    float* qf    = out;
    float* oidx  = qf + (size_t)M * CCH;
    float* ow    = oidx + (size_t)M * 8;
    float* accum = ow + (size_t)M * 8;

    init_kernel   <<<(OCC_SIZE + 255) / 256, 256, 0, stream>>>(shift, occ, accum, N);
    shift_kernel  <<<(N + 255) / 256,        256, 0, stream>>>(coords, shift, N);
    scatter_kernel<<<(N + 255) / 256,        256, 0, stream>>>(coords, shift, occ, N);
    query_kernel  <<<(M + 7) / 8,            256, 0, stream>>>(feat, (const float4*)qp,
                                                               shift, occ,
                                                               qf, oidx, ow, accum, M);
}